// GCN_10316511445242
// MI455X (gfx1250) — compile-verified
//
#include <hip/hip_runtime.h>
#include <math.h>
#include <stdint.h>

#define NN 100000
#define NE 1600000

typedef float v2f __attribute__((ext_vector_type(2)));
typedef float v8f __attribute__((ext_vector_type(8)));

// ---------------- utility kernels ----------------

__global__ void zero_kernel(float* __restrict__ p, long long n) {
  long long i = (long long)blockIdx.x * blockDim.x + threadIdx.x;
  long long stride = (long long)gridDim.x * blockDim.x;
  for (; i < n; i += stride) p[i] = 0.0f;
}

__global__ void degree_kernel(const int* __restrict__ src, const int* __restrict__ dst,
                              float* __restrict__ degs, float* __restrict__ degd) {
  int i = blockIdx.x * blockDim.x + threadIdx.x;
  if (i < NE) {
    atomicAdd(&degs[src[i]], 1.0f);
    atomicAdd(&degd[dst[i]], 1.0f);
  }
}

__global__ void norm_kernel(float* __restrict__ p, int n) {
  int i = blockIdx.x * blockDim.x + threadIdx.x;
  if (i < n) p[i] = 1.0f / sqrtf(fmaxf(p[i], 1.0f));
}

// ---------------- fused (norm_src * X) @ W via fp32 WMMA ----------------
// One wave computes one 16-row M tile across all DOUT columns.
// W staged through LDS in <=32KB chunks with GLOBAL_LOAD_ASYNC_TO_LDS_B128
// (CDNA5 async path, tracked by ASYNCcnt).

template <int DIN, int DOUT>
__global__ __launch_bounds__(256) void gemm_norm_wmma(
    const float* __restrict__ X, const float* __restrict__ nrm_src,
    const float* __restrict__ W, float* __restrict__ T) {
  constexpr int NT = DOUT / 16;                                   // N tiles
  constexpr int KC = (DIN * DOUT * 4 <= 32768) ? DIN : (8192 / DOUT);
  __shared__ alignas(16) float Wlds[KC * DOUT];

  const int tid   = threadIdx.x;
  const int lane  = tid & 31;
  const int wave  = tid >> 5;
  const int mtile = blockIdx.x * 8 + wave;
  const bool valid = (mtile * 16) < NN;                           // wave-uniform
  const int mrow  = mtile * 16 + (lane & 15);
  const int koff  = (lane >> 4) << 1;                             // 0 or 2
  const int ncol  = lane & 15;

  float nrm = 0.0f;
  const float* xrow = X;
  if (valid) {
    nrm  = nrm_src[mrow];
    xrow = X + (long long)mrow * DIN;
  }

  v8f acc[NT];
#pragma unroll
  for (int t = 0; t < NT; ++t)
    acc[t] = (v8f){0.f, 0.f, 0.f, 0.f, 0.f, 0.f, 0.f, 0.f};

  for (int k0 = 0; k0 < DIN; k0 += KC) {
    // --- async stage W[k0 : k0+KC, :] into LDS, 16B per lane per op ---
    {
      const float* gbase = W + k0 * DOUT;
      for (int i = tid; i < (KC * DOUT) / 4; i += 256) {
        const unsigned lds_off = (unsigned)(uintptr_t)(&Wlds[i * 4]); // low 32b = LDS byte offset
        const float* gp = gbase + i * 4;
        asm volatile("global_load_async_to_lds_b128 %0, %1, off"
                     :
                     : "v"(lds_off), "v"(gp)
                     : "memory");
      }
      asm volatile("s_wait_asynccnt 0" ::: "memory");
    }
    __syncthreads();

    if (valid) {
      for (int k = 0; k < KC; k += 4) {
        const v2f av = *reinterpret_cast<const v2f*>(xrow + k0 + k + koff);
        v2f a;                                                     // A: 16x4 fp32
        a.x = av.x * nrm;
        a.y = av.y * nrm;
#pragma unroll
        for (int t = 0; t < NT; ++t) {
          v2f b;                                                   // B: 4x16 fp32
          b.x = Wlds[(k + koff)     * DOUT + t * 16 + ncol];
          b.y = Wlds[(k + koff + 1) * DOUT + t * 16 + ncol];
          acc[t] = __builtin_amdgcn_wmma_f32_16x16x4_f32(
              false, a, false, b, (short)0, acc[t], false, false);
        }
      }
    }
    __syncthreads();
  }

  if (valid) {
    const int rbase = mtile * 16 + ((lane >> 4) << 3);            // +8 for lanes 16-31
#pragma unroll
    for (int t = 0; t < NT; ++t) {
#pragma unroll
      for (int r = 0; r < 8; ++r) {
        T[(long long)(rbase + r) * DOUT + t * 16 + ncol] = acc[t][r];
      }
    }
  }
}

// ---------------- SpMM aggregation: agg[dst] += T[src] ----------------

template <int D>
__global__ void agg_kernel(const float* __restrict__ T, const int* __restrict__ src,
                           const int* __restrict__ dst, float* __restrict__ agg) {
  long long idx = (long long)blockIdx.x * blockDim.x + threadIdx.x;
  const long long total = (long long)NE * D;
  const long long stride = (long long)gridDim.x * blockDim.x;
  for (; idx < total; idx += stride) {
    const int e = (int)(idx >> __builtin_ctz(D));   // D is power of two
    const int f = (int)(idx & (D - 1));
    const int s = src[e];
    const int d = dst[e];
    atomicAdd(&agg[(long long)d * D + f], T[(long long)s * D + f]);
  }
}

// ---------------- epilogue: h = relu(agg*norm_dst + b + sb) in place ----------------

template <int D>
__global__ void bias_relu_kernel(float* __restrict__ h, const float* __restrict__ norm_dst,
                                 const float* __restrict__ b, const float* __restrict__ sb) {
  long long idx = (long long)blockIdx.x * blockDim.x + threadIdx.x;
  const long long total = (long long)NN * D;
  const long long stride = (long long)gridDim.x * blockDim.x;
  const float sbv = sb[0];
  for (; idx < total; idx += stride) {
    const int n = (int)(idx >> __builtin_ctz(D));
    const int f = (int)(idx & (D - 1));
    const float v = h[idx] * norm_dst[n] + b[f] + sbv;
    h[idx] = fmaxf(v, 0.0f);
  }
}

// ---------------- final layer epilogue + softmax over 16 classes ----------------

__global__ void final_softmax_kernel(float* __restrict__ out, const float* __restrict__ norm_dst,
                                     const float* __restrict__ b, const float* __restrict__ sb) {
  const int n = blockIdx.x * blockDim.x + threadIdx.x;
  if (n >= NN) return;
  const float nd = norm_dst[n];
  const float sbv = sb[0];
  float v[16];
  float m = -INFINITY;
#pragma unroll
  for (int f = 0; f < 16; ++f) {
    v[f] = fmaxf(out[(long long)n * 16 + f] * nd + b[f] + sbv, 0.0f);
    m = fmaxf(m, v[f]);
  }
  float s = 0.0f;
#pragma unroll
  for (int f = 0; f < 16; ++f) { v[f] = expf(v[f] - m); s += v[f]; }
  const float inv = 1.0f / s;
#pragma unroll
  for (int f = 0; f < 16; ++f) out[(long long)n * 16 + f] = v[f] * inv;
}

// ---------------- host-side orchestration ----------------

static inline unsigned nblk(long long n, int t) { return (unsigned)((n + t - 1) / t); }

extern "C" void kernel_launch(void* const* d_in, const int* in_sizes, int n_in,
                              void* d_out, int out_size, void* d_ws, size_t ws_size,
                              hipStream_t stream) {
  const float* x   = (const float*)d_in[0];
  const int*   src = (const int*)d_in[1];
  const int*   dst = (const int*)d_in[2];
  const float* W0 = (const float*)d_in[3];  const float* b0 = (const float*)d_in[4];  const float* sb0 = (const float*)d_in[5];
  const float* W1 = (const float*)d_in[6];  const float* b1 = (const float*)d_in[7];  const float* sb1 = (const float*)d_in[8];
  const float* W2 = (const float*)d_in[9];  const float* b2 = (const float*)d_in[10]; const float* sb2 = (const float*)d_in[11];
  const float* W3 = (const float*)d_in[12]; const float* b3 = (const float*)d_in[13]; const float* sb3 = (const float*)d_in[14];
  float* out = (float*)d_out;

  // workspace carve-up (~129 MB total)
  char* ws = (char*)d_ws;
  size_t off = 0;
  auto carve = [&](size_t bytes) -> float* {
    float* p = (float*)(ws + off);
    off = (off + bytes + 255) & ~(size_t)255;
    return p;
  };
  float* norm_src = carve((size_t)NN * 4);
  float* norm_dst = carve((size_t)NN * 4);
  float* bufT = carve((size_t)NN * 128 * 4);   // transformed features (pre-aggregation)
  float* bufA = carve((size_t)NN * 128 * 4);   // ping
  float* bufB = carve((size_t)NN * 64  * 4);   // pong

  const int gemm_grid = (NN / 16 + 7) / 8;     // 6250 M-tiles, 8 waves/block

  // degrees -> norms
  zero_kernel<<<nblk(NN, 256), 256, 0, stream>>>(norm_src, NN);
  zero_kernel<<<nblk(NN, 256), 256, 0, stream>>>(norm_dst, NN);
  degree_kernel<<<nblk(NE, 256), 256, 0, stream>>>(src, dst, norm_src, norm_dst);
  norm_kernel<<<nblk(NN, 256), 256, 0, stream>>>(norm_src, NN);
  norm_kernel<<<nblk(NN, 256), 256, 0, stream>>>(norm_dst, NN);

  // layer 0: 256 -> 128
  gemm_norm_wmma<256, 128><<<gemm_grid, 256, 0, stream>>>(x, norm_src, W0, bufT);
  zero_kernel<<<nblk((long long)NN * 128, 256), 256, 0, stream>>>(bufA, (long long)NN * 128);
  agg_kernel<128><<<nblk((long long)NE * 128, 256), 256, 0, stream>>>(bufT, src, dst, bufA);
  bias_relu_kernel<128><<<nblk((long long)NN * 128, 256), 256, 0, stream>>>(bufA, norm_dst, b0, sb0);

  // layer 1: 128 -> 64
  gemm_norm_wmma<128, 64><<<gemm_grid, 256, 0, stream>>>(bufA, norm_src, W1, bufT);
  zero_kernel<<<nblk((long long)NN * 64, 256), 256, 0, stream>>>(bufB, (long long)NN * 64);
  agg_kernel<64><<<nblk((long long)NE * 64, 256), 256, 0, stream>>>(bufT, src, dst, bufB);
  bias_relu_kernel<64><<<nblk((long long)NN * 64, 256), 256, 0, stream>>>(bufB, norm_dst, b1, sb1);

  // layer 2: 64 -> 32
  gemm_norm_wmma<64, 32><<<gemm_grid, 256, 0, stream>>>(bufB, norm_src, W2, bufT);
  zero_kernel<<<nblk((long long)NN * 32, 256), 256, 0, stream>>>(bufA, (long long)NN * 32);
  agg_kernel<32><<<nblk((long long)NE * 32, 256), 256, 0, stream>>>(bufT, src, dst, bufA);
  bias_relu_kernel<32><<<nblk((long long)NN * 32, 256), 256, 0, stream>>>(bufA, norm_dst, b2, sb2);

  // layer 3: 32 -> 16, aggregate straight into d_out, then fused relu+softmax in place
  gemm_norm_wmma<32, 16><<<gemm_grid, 256, 0, stream>>>(bufA, norm_src, W3, bufT);
  zero_kernel<<<nblk((long long)NN * 16, 256), 256, 0, stream>>>(out, (long long)NN * 16);
  agg_kernel<16><<<nblk((long long)NE * 16, 256), 256, 0, stream>>>(bufT, src, dst, out);
  final_softmax_kernel<<<nblk(NN, 256), 256, 0, stream>>>(out, norm_dst, b3, sb3);
}